// Attn_55293408969033
// MI455X (gfx1250) — compile-verified
//
#include <hip/hip_runtime.h>

// ============================================================================
// ESIM-style bidirectional masked attention for MI455X (gfx1250, wave32).
//
// Roofline: 60 GFLOP fp32, ~315 MB mandatory HBM traffic (~14 us @ 23.3 TB/s).
// Scores feed exp(), so matmuls stay fp32 via native CDNA5 WMMA
// (V_WMMA_F32_16X16X4_F32). Operands for the big GEMMs are staged into LDS by
// the Tensor Data Mover (tensor_load_to_lds + s_wait_tensorcnt), double
// buffered so the DMA overlaps the WMMA stream. Intermediates (attn 64MB +
// attnT 64MB in d_ws) largely live in the 192MB L2.
// ============================================================================

typedef __attribute__((ext_vector_type(2))) float        v2f;
typedef __attribute__((ext_vector_type(8))) float        v8f;
typedef __attribute__((ext_vector_type(4))) unsigned int u32x4;
typedef __attribute__((ext_vector_type(4))) int          i32x4;
typedef __attribute__((ext_vector_type(8))) int          i32x8;

#define BATCH 64
#define LSEQ  512
#define DDIM  600
#define EPSR  1e-13f

#define KC 32   // K-chunk staged per TDM transfer (dwords per tile row)
#define KP 36   // LDS row stride in floats (KC + 4 pad dwords, conflict-free)

#if __has_builtin(__builtin_amdgcn_tensor_load_to_lds)
#define HAVE_TDM 1
#else
#define HAVE_TDM 0
#endif

// LDS byte offset of a shared-memory object (generic pointer low 32 bits carry
// the LDS segment offset on AMDGCN).
__device__ __forceinline__ unsigned lds_addr_of(const void* p) {
    return (unsigned)(uintptr_t)p;
}

// ---------------------------------------------------------------------------
// Stage a [rows x KC] fp32 tile from a row-major (stride DDIM) global slab
// into LDS rows of stride KP, zero-filling dim0 elements past `valid0`.
//  - TDM path: wave 0 issues one tensor_load_to_lds (async, TENSORcnt).
//    D# is relative to the tile start: tensor_dim0 = valid0 gives hardware
//    zero-fill of the padded columns; pad fields produce the KP stride.
//  - Fallback: whole-block cooperative synchronous copy.
// ---------------------------------------------------------------------------
__device__ __forceinline__ void stage_issue(float* dst, const float* src,
                                            int valid0, int rem1, int rows) {
#if HAVE_TDM
    if (threadIdx.x < 32) {
        unsigned long long ga = (unsigned long long)(uintptr_t)src;
        u32x4 g0;
        g0[0] = 1u;                                   // count=1, user mode
        g0[1] = lds_addr_of(dst);                     // lds_addr
        g0[2] = (unsigned)(ga & 0xFFFFFFFFu);         // global_addr lo
        g0[3] = (unsigned)((ga >> 32) & 0x1FFFFFFu)   // global_addr hi
              | (2u << 30);                           // type=2 (image/tensor)
        i32x8 g1;
        g1[0] = (2 << 16)      // data_size = 4B
              | (1 << 20)      // pad_enable
              | (4 << 22)      // pad_interval: 1<<4 * 8B = 32 dwords (= KC)
              | (3 << 25);     // pad_amount: 4 dwords -> row stride KP=36
        g1[1] = (valid0 & 0xFFFF) << 16;                              // dim0 lo
        g1[2] = ((valid0 >> 16) & 0xFFFF) | ((rem1 & 0xFFFF) << 16);  // dim0 hi | dim1 lo
        g1[3] = ((rem1 >> 16) & 0xFFFF) | (KC << 16);                 // dim1 hi | tile_dim0
        g1[4] = rows & 0xFFFF;                                        // tile_dim1 (tile_dim2=0)
        g1[5] = DDIM;                                                 // dim0 stride lo
        g1[6] = 0;                                                    // stride hi | stride1 lo
        g1[7] = 0;
        i32x4 z4 = (i32x4)0;
#if __clang_major__ >= 23
        i32x8 z8 = (i32x8)0;
        __builtin_amdgcn_tensor_load_to_lds(g0, g1, z4, z4, z8, 0);
#else
        __builtin_amdgcn_tensor_load_to_lds(g0, g1, z4, z4, 0);
#endif
    }
#else
    (void)rem1;
    int total = rows * (KC / 4);
    for (int i = threadIdx.x; i < total; i += 256) {
        int rr = i / (KC / 4);
        int x4 = (i % (KC / 4)) * 4;
        const float* s = src + (size_t)rr * DDIM + x4;
        float4 v;
        v.x = (x4 + 0 < valid0) ? s[0] : 0.0f;
        v.y = (x4 + 1 < valid0) ? s[1] : 0.0f;
        v.z = (x4 + 2 < valid0) ? s[2] : 0.0f;
        v.w = (x4 + 3 < valid0) ? s[3] : 0.0f;
        *(float4*)(dst + rr * KP + x4) = v;
    }
#endif
}

__device__ __forceinline__ void stage_wait() {
#if HAVE_TDM
    if (threadIdx.x < 32) {
#if __has_builtin(__builtin_amdgcn_s_wait_tensorcnt)
        __builtin_amdgcn_s_wait_tensorcnt(0);
#else
        asm volatile("s_wait_tensorcnt 0x0" ::: "memory");
#endif
    }
#endif
}

// WMMA f32 16x16x4 operand layout (ISA 7.12.2):
//   lane in [0,16):  row M=lane,    v0=K+0, v1=K+1
//   lane in [16,32): row M=lane-16, v0=K+2, v1=K+3   (B mirrors with N for M)

// ---------------------------------------------------------------------------
// Kernel 1: scores[b,p,h] = sum_d P[b,p,d] * H[b,h,d]
// Block = 8 waves = 128(p) x 64(h) tile; wave = 32x32 (2x2 WMMA blocking).
// P/H K-chunks staged in LDS by TDM, double buffered. The 32x32 result tile
// is stored to attn[b,p,h] AND attnT[b,h,p] so both softmaxes are row-wise.
// ---------------------------------------------------------------------------
__global__ __launch_bounds__(256) void gemm_scores(
    const float* __restrict__ P,
    const float* __restrict__ H,
    float* __restrict__ attn,
    float* __restrict__ attnT)
{
    __shared__ float Ps[2][128 * KP];
    __shared__ float Hs[2][64 * KP];

    int blk  = blockIdx.x;               // 64 * 32 blocks
    int b    = blk >> 5;
    int t    = blk & 31;
    int pblk = (t >> 3) * 128;           // 4 p-blocks
    int hblk = (t & 7) * 64;             // 8 h-blocks

    int wave = threadIdx.x >> 5;
    int lane = threadIdx.x & 31;
    int half = lane >> 4;
    int lp   = lane & 15;
    int wp   = wave >> 1;                // 0..3
    int wh   = wave & 1;                 // 0..1

    const float* Pb = P + (size_t)b * LSEQ * DDIM;
    const float* Hb = H + (size_t)b * LSEQ * DDIM;

    const int NCH = (DDIM + KC - 1) / KC;    // 19 chunks (last zero-padded)

    stage_issue(&Ps[0][0], Pb + (size_t)pblk * DDIM, DDIM, LSEQ - pblk, 128);
    stage_issue(&Hs[0][0], Hb + (size_t)hblk * DDIM, DDIM, LSEQ - hblk, 64);

    v8f c00 = {}, c01 = {}, c10 = {}, c11 = {};

    for (int ch = 0; ch < NCH; ++ch) {
        int buf = ch & 1;
        stage_wait();
        __syncthreads();
        if (ch + 1 < NCH) {
            int k0 = (ch + 1) * KC;
            stage_issue(&Ps[buf ^ 1][0], Pb + (size_t)pblk * DDIM + k0,
                        DDIM - k0, LSEQ - pblk, 128);
            stage_issue(&Hs[buf ^ 1][0], Hb + (size_t)hblk * DDIM + k0,
                        DDIM - k0, LSEQ - hblk, 64);
        }
        const float* pA = &Ps[buf][(32 * wp + lp) * KP + 2 * half];
        const float* pB = &Hs[buf][(32 * wh + lp) * KP + 2 * half];
        #pragma unroll
        for (int kk = 0; kk < KC; kk += 4) {
            v2f a0 = *(const v2f*)(pA + kk);
            v2f a1 = *(const v2f*)(pA + 16 * KP + kk);
            v2f b0 = *(const v2f*)(pB + kk);
            v2f b1 = *(const v2f*)(pB + 16 * KP + kk);
            c00 = __builtin_amdgcn_wmma_f32_16x16x4_f32(false, a0, false, b0, (short)0, c00, false, false);
            c01 = __builtin_amdgcn_wmma_f32_16x16x4_f32(false, a0, false, b1, (short)0, c01, false, false);
            c10 = __builtin_amdgcn_wmma_f32_16x16x4_f32(false, a1, false, b0, (short)0, c10, false, false);
            c11 = __builtin_amdgcn_wmma_f32_16x16x4_f32(false, a1, false, b1, (short)0, c11, false, false);
        }
        __syncthreads();
    }

    int p0 = pblk + 32 * wp;
    int h0 = hblk + 32 * wh;
    float* A  = attn  + (size_t)b * LSEQ * LSEQ;
    float* AT = attnT + (size_t)b * LSEQ * LSEQ;

    // C/D layout: VGPR i, lanes<16 -> M=i, N=lane; lanes>=16 -> M=8+i, N=lane-16
    #pragma unroll
    for (int mi = 0; mi < 2; ++mi) {
        #pragma unroll
        for (int ni = 0; ni < 2; ++ni) {
            v8f c = (mi == 0) ? (ni == 0 ? c00 : c01) : (ni == 0 ? c10 : c11);
            int pm = p0 + 16 * mi;
            int hn = h0 + 16 * ni;
            #pragma unroll
            for (int i = 0; i < 8; ++i) {
                int row = pm + i + 8 * half;     // p index
                int col = hn + lp;               // h index
                float v = c[i];
                A [(size_t)row * LSEQ + col] = v;
                AT[(size_t)col * LSEQ + row] = v;
            }
        }
    }
}

// ---------------------------------------------------------------------------
// Kernel 2: in-place masked softmax over the last axis of S[b, r, :].
// Reference algebra: r = softmax(s*m)*m; out = r/(sum(r)+EPS)
//   => out_j = e_j*m_j / (sum(e*m) + EPS*sum(e)),  e_j = exp(s_j*m_j - max)
// ---------------------------------------------------------------------------
__global__ __launch_bounds__(256) void row_softmax_inplace(
    float* __restrict__ S,
    const float* __restrict__ mask)
{
    int row = blockIdx.x;                 // b*LSEQ + r
    int b   = row / LSEQ;
    float*       s  = S    + (size_t)row * LSEQ;
    const float* mk = mask + (size_t)b   * LSEQ;

    int tid = threadIdx.x;
    float2 sv = *(const float2*)(s  + 2 * tid);
    float2 mv = *(const float2*)(mk + 2 * tid);
    float t0 = sv.x * mv.x;
    float t1 = sv.y * mv.y;

    __shared__ float red[256];

    red[tid] = fmaxf(t0, t1);
    __syncthreads();
    #pragma unroll
    for (int off = 128; off > 0; off >>= 1) {
        if (tid < off) red[tid] = fmaxf(red[tid], red[tid + off]);
        __syncthreads();
    }
    float M = red[0];
    __syncthreads();

    float e0 = __expf(t0 - M);
    float e1 = __expf(t1 - M);

    red[tid] = e0 + e1;
    __syncthreads();
    #pragma unroll
    for (int off = 128; off > 0; off >>= 1) {
        if (tid < off) red[tid] += red[tid + off];
        __syncthreads();
    }
    float denom = red[0];
    __syncthreads();

    red[tid] = e0 * mv.x + e1 * mv.y;
    __syncthreads();
    #pragma unroll
    for (int off = 128; off > 0; off >>= 1) {
        if (tid < off) red[tid] += red[tid + off];
        __syncthreads();
    }
    float msum = red[0];
    __syncthreads();

    float inv = 1.0f / (msum + EPSR * denom);
    float2 o;
    o.x = e0 * mv.x * inv;
    o.y = e1 * mv.y * inv;
    *(float2*)(s + 2 * tid) = o;
}

// ---------------------------------------------------------------------------
// Kernel 3: out[b,m,d] = mask[b,m] * sum_k A[b,m,k] * V[b,k,d]
// Block = 8 waves = 256(m) rows x one 32-wide d-tile. A rows stream from
// global (contiguous, L2-resident); the strided V tile is TDM-staged in LDS
// (zero-filled past d=600 by tensor_dim0, so no clamps needed).
// ---------------------------------------------------------------------------
__global__ __launch_bounds__(256) void gemm_weighted(
    const float* __restrict__ Asm,     // [B,512,512]
    const float* __restrict__ V,       // [B,512,600]
    const float* __restrict__ mask,    // [B,512] mask along M
    float* __restrict__ out)           // [B,512,600]
{
    __shared__ float Vs[2][KC * KP];   // 32 k-rows x 32(+4 pad) d-cols

    const int NT = 19;                 // ceil(600/32) d-tiles
    int blk = blockIdx.x;              // 64 * 2 * 19 blocks
    int b   = blk / (2 * NT);
    int r   = blk % (2 * NT);
    int mblk = (r / NT) * 256;
    int n0   = (r % NT) * 32;

    int wave = threadIdx.x >> 5;
    int lane = threadIdx.x & 31;
    int half = lane >> 4;
    int lp   = lane & 15;
    int m0   = mblk + wave * 32;

    const float* Ab = Asm + (size_t)b * LSEQ * LSEQ;
    const float* Vb = V   + (size_t)b * LSEQ * DDIM;
    const float* a0p = Ab + (size_t)(m0 + lp) * LSEQ + 2 * half;
    const float* a1p = a0p + (size_t)16 * LSEQ;

    const int NCH = LSEQ / KC;         // 16 K-chunks

    stage_issue(&Vs[0][0], Vb + n0, DDIM - n0, LSEQ, KC);

    v8f c00 = {}, c01 = {}, c10 = {}, c11 = {};

    for (int ch = 0; ch < NCH; ++ch) {
        int buf = ch & 1;
        int k0  = ch * KC;
        stage_wait();
        __syncthreads();
        if (ch + 1 < NCH) {
            stage_issue(&Vs[buf ^ 1][0], Vb + (size_t)(k0 + KC) * DDIM + n0,
                        DDIM - n0, LSEQ - (k0 + KC), KC);
        }
        const float* pB = &Vs[buf][0];
        #pragma unroll
        for (int kk = 0; kk < KC; kk += 4) {
            v2f a0 = *(const v2f*)(a0p + k0 + kk);
            v2f a1 = *(const v2f*)(a1p + k0 + kk);
            int rr = kk + 2 * half;
            v2f b0, b1;
            b0.x = pB[(rr    ) * KP + lp];
            b0.y = pB[(rr + 1) * KP + lp];
            b1.x = pB[(rr    ) * KP + lp + 16];
            b1.y = pB[(rr + 1) * KP + lp + 16];
            c00 = __builtin_amdgcn_wmma_f32_16x16x4_f32(false, a0, false, b0, (short)0, c00, false, false);
            c01 = __builtin_amdgcn_wmma_f32_16x16x4_f32(false, a0, false, b1, (short)0, c01, false, false);
            c10 = __builtin_amdgcn_wmma_f32_16x16x4_f32(false, a1, false, b0, (short)0, c10, false, false);
            c11 = __builtin_amdgcn_wmma_f32_16x16x4_f32(false, a1, false, b1, (short)0, c11, false, false);
        }
        __syncthreads();
    }

    const float* mrow = mask + (size_t)b * LSEQ;
    float* ob = out + (size_t)b * LSEQ * DDIM;

    #pragma unroll
    for (int mi = 0; mi < 2; ++mi) {
        #pragma unroll
        for (int ni = 0; ni < 2; ++ni) {
            v8f c = (mi == 0) ? (ni == 0 ? c00 : c01) : (ni == 0 ? c10 : c11);
            int mbase = m0 + 16 * mi;
            int d     = n0 + 16 * ni + lp;
            if (d < DDIM) {
                #pragma unroll
                for (int i = 0; i < 8; ++i) {
                    int m = mbase + i + 8 * half;
                    ob[(size_t)m * DDIM + d] = c[i] * mrow[m];
                }
            }
        }
    }
}

// ---------------------------------------------------------------------------
// Host-side launcher.
// d_in: [0]=encoded_premise (64,512,600) f32, [1]=premise_mask (64,512) f32,
//       [2]=encoded_hypothesis (64,512,600) f32, [3]=hypothesis_mask (64,512)
// d_out: weighted_premise (64,512,600) ++ weighted_hypothesis (64,512,600)
// d_ws : attn (64MB) ++ attnT (64MB) = 128MB
// ---------------------------------------------------------------------------
extern "C" void kernel_launch(void* const* d_in, const int* in_sizes, int n_in,
                              void* d_out, int out_size, void* d_ws, size_t ws_size,
                              hipStream_t stream) {
    const float* P     = (const float*)d_in[0];
    const float* pmask = (const float*)d_in[1];
    const float* H     = (const float*)d_in[2];
    const float* hmask = (const float*)d_in[3];

    float* attn  = (float*)d_ws;
    float* attnT = attn + (size_t)BATCH * LSEQ * LSEQ;

    float* out_p = (float*)d_out;
    float* out_h = out_p + (size_t)BATCH * LSEQ * DDIM;

    // 1) scores, dual layout: 64 batches x (4 p-blocks x 8 h-blocks)
    gemm_scores<<<BATCH * 32, 256, 0, stream>>>(P, H, attn, attnT);

    // 2) masked softmax rows (h axis on attn, p axis on attnT), in place
    row_softmax_inplace<<<BATCH * LSEQ, 256, 0, stream>>>(attn,  hmask);
    row_softmax_inplace<<<BATCH * LSEQ, 256, 0, stream>>>(attnT, pmask);

    // 3) weighted sums: 64 x 2 m-blocks x 19 d-tiles
    gemm_weighted<<<BATCH * 2 * 19, 256, 0, stream>>>(attn,  H, pmask, out_p);
    gemm_weighted<<<BATCH * 2 * 19, 256, 0, stream>>>(attnT, P, hmask, out_h);
}